// KDELayer_60129542697
// MI455X (gfx1250) — compile-verified
//
#include <hip/hip_runtime.h>
#include <cmath>

// ---------------------------------------------------------------------------
// KDE layer: out[b,m] = (coef/N) * sum_n exp(-0.5*||p_m - x_{b,n}||^2 / h^2)
// B=128, N=2048, d=2, M=512.
//
// Distance math on the matrix pipe via V_WMMA_F32_16X16X4_F32 (K=4 exact):
//   A[m] = (c*|p|^2, c*p.x, c*p.y, c),  B[:,n] = (1, -2x, -2y, |x|^2)
// with c = -0.5*log2(e)/h^2, so D is directly the BASE-2 exponent and the
// kernel value is a single native v_exp_f32 (exp2) — no per-element scaling
// mul. exp2 runs on the trans pipe and co-executes with the XDL WMMA.
// ---------------------------------------------------------------------------

typedef float v2f __attribute__((ext_vector_type(2)));
typedef float v8f __attribute__((ext_vector_type(8)));

#define B_DIM  128
#define N_SAMP 2048
#define M_PTS  512

__global__ __launch_bounds__(32)
void kde_wmma_kernel(const float* __restrict__ inputs,      // (B, N, 2)
                     const float* __restrict__ points,      // (M, 2)
                     float* __restrict__ out,               // (B, M)
                     float cneg,                            // -0.5*log2(e)/h^2
                     float scale)                           // coef / N
{
    const int lane = threadIdx.x;     // 0..31, wave32
    const int idx  = lane & 15;       // position within half-wave
    const int hi   = lane >> 4;       // half-wave selects K pair / D rows
    const int mtile = blockIdx.x;     // 0..M/16-1
    const int b     = blockIdx.y;     // 0..B-1

    // --- A matrix (constant over the N loop): 16x4 f32, 2 VGPRs/lane -------
    // lanes 0-15: (A[m][0], A[m][1]); lanes 16-31: (A[m][2], A[m][3]); m=idx
    const float2* pts = (const float2*)points;
    float2 p = pts[mtile * 16 + idx];
    v2f amat;
    amat.x = hi ? (cneg * p.y) : (cneg * (p.x * p.x + p.y * p.y));
    amat.y = hi ? (cneg)       : (cneg * p.x);

    const float2* xb = (const float2*)inputs + (size_t)b * N_SAMP;

    v8f acc0 = {0.f,0.f,0.f,0.f,0.f,0.f,0.f,0.f};
    v8f acc1 = {0.f,0.f,0.f,0.f,0.f,0.f,0.f,0.f};

    for (int n0 = 0; n0 < N_SAMP; n0 += 32) {
        // Two independent 16-sample chunks per iteration for ILP.
        float2 s0 = xb[n0 + idx];
        float2 s1 = xb[n0 + 16 + idx];

        // B matrix 4x16 f32, 2 VGPRs/lane; column n = idx:
        // lanes 0-15: (B[0][n], B[1][n]); lanes 16-31: (B[2][n], B[3][n])
        v2f b0, b1;
        b0.x = hi ? (-2.0f * s0.y) : 1.0f;
        b0.y = hi ? (s0.x * s0.x + s0.y * s0.y) : (-2.0f * s0.x);
        b1.x = hi ? (-2.0f * s1.y) : 1.0f;
        b1.y = hi ? (s1.x * s1.x + s1.y * s1.y) : (-2.0f * s1.x);

        v8f z = {0.f,0.f,0.f,0.f,0.f,0.f,0.f,0.f};
        // D = A x B + 0 : 16x16 tile of base-2 exponents
        v8f d0 = __builtin_amdgcn_wmma_f32_16x16x4_f32(
                     false, amat, false, b0, (short)0, z, false, false);
        v8f d1 = __builtin_amdgcn_wmma_f32_16x16x4_f32(
                     false, amat, false, b1, (short)0, z, false, false);

#pragma unroll
        for (int i = 0; i < 8; ++i) {
            acc0[i] += __builtin_amdgcn_exp2f(d0[i]);   // raw v_exp_f32
            acc1[i] += __builtin_amdgcn_exp2f(d1[i]);
        }
    }

    // --- Reduce each D row (16 lanes within a half-wave) -------------------
    // D layout: VGPR i holds row m=i (lanes 0-15) and m=i+8 (lanes 16-31).
#pragma unroll
    for (int i = 0; i < 8; ++i) {
        float v = acc0[i] + acc1[i];
        v += __shfl_xor(v, 1, 32);
        v += __shfl_xor(v, 2, 32);
        v += __shfl_xor(v, 4, 32);
        v += __shfl_xor(v, 8, 32);   // xor masks < 16 stay within each half
        if (idx == 0) {
            int m = mtile * 16 + i + (hi ? 8 : 0);
            out[(size_t)b * M_PTS + m] = scale * v;
        }
    }
}

extern "C" void kernel_launch(void* const* d_in, const int* in_sizes, int n_in,
                              void* d_out, int out_size, void* d_ws, size_t ws_size,
                              hipStream_t stream) {
    (void)in_sizes; (void)n_in; (void)out_size; (void)d_ws; (void)ws_size;

    const float* inputs = (const float*)d_in[0];   // (128, 2048, 2) f32
    const float* points = (const float*)d_in[1];   // (512, 2) f32
    float* out = (float*)d_out;                    // (128, 512) f32

    // Silverman bandwidth, computed in double on the host.
    const double d = 2.0;
    const double N = (double)N_SAMP;
    const double PI    = 3.14159265358979323846;
    const double LOG2E = 1.44269504088896340736;   // fold into WMMA constant
    double h    = pow(4.0 / (d + 2.0), 1.0 / (d + 4.0)) * pow(N, -1.0 / (d + 4.0));
    double coef = 1.0 / pow(2.0 * PI * h * h, d / 2.0);

    float cneg  = (float)(-0.5 * LOG2E / (h * h));
    float scale = (float)(coef / N);

    dim3 grid(M_PTS / 16, B_DIM);   // 32 m-tiles x 128 batches = 4096 waves
    kde_wmma_kernel<<<grid, 32, 0, stream>>>(inputs, points, out, cneg, scale);
}